// BNOS_21569325760982
// MI455X (gfx1250) — compile-verified
//
#include <hip/hip_runtime.h>
#include <math.h>
#include <stdint.h>

// ---------------------------------------------------------------------------
// Born-series neural operator on gfx1250 (MI455X).
// - All dense contractions: V_WMMA_F32_16X16X4_F32 (exact f32 WMMA path).
// - 2D rFFT/irFFT as DFT GEMMs with precomputed cos/sin bases.
// - GEMM tile staging: GLOBAL_LOAD_ASYNC_TO_LDS_B128 + s_wait_asynccnt.
// - MLP weights pre-packed into WMMA B-fragment layout (no branchy loads).
// ---------------------------------------------------------------------------

typedef __attribute__((ext_vector_type(2))) float v2f;
typedef __attribute__((ext_vector_type(8))) float v8f;

#define NB    4
#define HH    256
#define WWI   256
#define CC    8            // WIDTH
#define FRQ   129          // W//2+1
#define FP    144          // freq padded to multiple of 16
#define PIX   (NB*HH*WWI)  // 262144 pixels
#define MFREQ (HH*FP)      // 36864 frequency-grid rows
#define SLAB  (HH*FP)      // per-(n,c) spectral slab elements
#define NSLAB (NB*CC)      // 32 slabs
#define MROWS (NB*CC*HH)   // 8192 rows for W-axis DFT GEMMs
#define ITERS 2
#define DEPTH 4
#define PKSTRIDE 1664      // packed floats per projection (1632 used)

__device__ __forceinline__ v8f wmma4(v2f a, v2f b, v8f c) {
  // (neg_a, A, neg_b, B, c_mod, C, reuse_a, reuse_b)
  return __builtin_amdgcn_wmma_f32_16x16x4_f32(false, a, false, b, (short)0, c,
                                               false, false);
}

__device__ __forceinline__ float geluf(float x) {
  // jax.nn.gelu approximate=True (tanh form)
  float x3 = x * x * x;
  return 0.5f * x * (1.0f + tanhf(0.7978845608028654f * (x + 0.044715f * x3)));
}

__device__ __forceinline__ void async_lds_b128(unsigned ldsOff, const float* g) {
  unsigned long long ga = (unsigned long long)(uintptr_t)g;
  asm volatile("global_load_async_to_lds_b128 %0, %1, off"
               :: "v"(ldsOff), "v"(ga) : "memory");
}
__device__ __forceinline__ void wait_async0() {
#if __has_builtin(__builtin_amdgcn_s_wait_asynccnt)
  __builtin_amdgcn_s_wait_asynccnt(0);
#else
  asm volatile("s_wait_asynccnt 0x0" ::: "memory");
#endif
}

// ---------------------------------------------------------------------------
// Basis matrices for DFT-as-GEMM.  Twiddle exactness: angle uses (i*j) mod 256.
// ---------------------------------------------------------------------------
__global__ void init_bases(float* Bc, float* Bns, float* Tc, float* Ts,
                           float* Tns, float* IBc, float* IBs) {
  int tid = blockIdx.x * blockDim.x + threadIdx.x;
  const float STEP = 6.283185307179586f / 256.0f;
  if (tid < HH * HH) {
    int g = tid >> 8, h = tid & 255;
    float ang = STEP * (float)((g * h) & 255);
    float c = cosf(ang), s = sinf(ang);
    Tc[tid] = c; Ts[tid] = s; Tns[tid] = -s;
  }
  if (tid < HH * FP) {
    int w = tid / FP, f = tid % FP;
    if (f < FRQ) {
      float ang = STEP * (float)((w * f) & 255);
      Bc[tid] = cosf(ang); Bns[tid] = -sinf(ang);
    } else { Bc[tid] = 0.f; Bns[tid] = 0.f; }
  }
  if (tid < FP * HH) {
    int f = tid / HH, w = tid & 255;
    if (f < FRQ) {
      float wgt = (f == 0 || f == 128) ? 1.f : 2.f;
      float ang = STEP * (float)((f * w) & 255);
      IBc[tid] =  wgt * cosf(ang) * (1.f / 65536.f);
      IBs[tid] = -wgt * sinf(ang) * (1.f / 65536.f);
    } else { IBc[tid] = 0.f; IBs[tid] = 0.f; }
  }
}

// k field [PIX][16]: sos | pml(4) | src(2) | w/255 | h/255 | zeros
__global__ void build_k(const float* __restrict__ sos,
                        const float* __restrict__ pml,
                        const float* __restrict__ src, float* __restrict__ kf) {
  int p = blockIdx.x * blockDim.x + threadIdx.x;
  if (p >= PIX) return;
  int w = p & 255, h = (p >> 8) & 255;
  float* o = kf + (size_t)p * 16;
  o[0] = sos[p];
  o[1] = pml[(size_t)p*4+0]; o[2] = pml[(size_t)p*4+1];
  o[3] = pml[(size_t)p*4+2]; o[4] = pml[(size_t)p*4+3];
  o[5] = src[(size_t)p*2+0]; o[6] = src[(size_t)p*2+1];
  o[7] = (float)w * (1.f/255.f); o[8] = (float)h * (1.f/255.f);
  #pragma unroll
  for (int i = 9; i < 16; ++i) o[i] = 0.f;
}

// frequency-grid field [MFREQ][16]: Kx=f (rfftfreq), Ky=g or g-256 (fftfreq)
__global__ void build_kq(float* __restrict__ kq) {
  int t = blockIdx.x * blockDim.x + threadIdx.x;
  if (t >= MFREQ) return;
  int g = t / FP, f = t % FP;
  float* o = kq + (size_t)t * 16;
  o[0] = (float)(f > 128 ? 128 : f);
  o[1] = (float)(g < 128 ? g : g - 256);
  #pragma unroll
  for (int i = 2; i < 16; ++i) o[i] = 0.f;
}

// ---------------------------------------------------------------------------
// Pack one projection's weights into WMMA B-fragment layout, zero-padded.
// Layout (floats): [0..255] w1 frags | [256..511] w2 frags |
//                  [512..1535] w3 frags (ntiles*256) |
//                  [1536..1551] b1 | [1552..1567] b2 | [1568..1631] b3 pad
// Fragment pair index = step*32 + lane; lane: n=lane&15, hi=lane>>4,
// kk = 4*step + 2*hi; pair = {w[kk][n], w[kk+1][n]}.
// ---------------------------------------------------------------------------
__global__ void pack_proj(const float* __restrict__ w1, const float* __restrict__ b1, int cin,
                          const float* __restrict__ w2, const float* __restrict__ b2,
                          const float* __restrict__ w3, const float* __restrict__ b3,
                          int cout, int ntiles, float* __restrict__ dst) {
  int t = threadIdx.x;                 // 256 threads, 1 block
  if (t < 128) {
    int s = t >> 5, lane = t & 31, n = lane & 15, hi = lane >> 4;
    int kk = 4 * s + 2 * hi;
    dst[t*2+0]     = (kk     < cin) ? w1[kk * 16 + n]       : 0.f;
    dst[t*2+1]     = (kk + 1 < cin) ? w1[(kk + 1) * 16 + n] : 0.f;
    dst[256+t*2+0] = w2[kk * 16 + n];
    dst[256+t*2+1] = w2[(kk + 1) * 16 + n];
  }
  for (int idx = t; idx < 128 * ntiles; idx += 256) {
    int tile = idx >> 7, r = idx & 127;
    int s = r >> 5, lane = r & 31, n = lane & 15, hi = lane >> 4;
    int kk = 4 * s + 2 * hi;
    int ng = tile * 16 + n;
    dst[512 + idx*2 + 0] = (ng < cout) ? w3[kk * cout + ng]       : 0.f;
    dst[512 + idx*2 + 1] = (ng < cout) ? w3[(kk + 1) * cout + ng] : 0.f;
  }
  if (t < 16) { dst[1536 + t] = b1[t]; dst[1552 + t] = b2[t]; }
  if (t < 64) { dst[1568 + t] = (t < cout) ? b3[t] : 0.f; }
}

// ---------------------------------------------------------------------------
// Per-row MLP:  Y = (gelu(gelu(X@W1+b1)@W2+b2)@W3+b3)*scale
// X [M][16] zero-padded; P = packed weights; one wave per 16-row tile.
// ---------------------------------------------------------------------------
#define PW 8
__global__ void proj_mlp(const float* __restrict__ X, int M,
                         const float* __restrict__ P, int ntiles,
                         float* __restrict__ Y, float scale) {
  __shared__ float sm_all[PW][16][17];
  int wave = threadIdx.x >> 5;
  int lane = threadIdx.x & 31;
  int n    = lane & 15;
  int hi   = lane >> 4;
  int row0 = (blockIdx.x * PW + wave) * 16;
  if (row0 >= M) return;
  float (*sm)[17] = sm_all[wave];
  const v2f* W1P = (const v2f*)P;            // 128 pairs
  const v2f* W2P = (const v2f*)(P + 256);
  const v2f* W3P = (const v2f*)(P + 512);
  const float* B1 = P + 1536;
  const float* B2 = P + 1552;
  const float* B3 = P + 1568;
  int coutPad = ntiles * 16;

  // layer 1
  v8f acc = {0,0,0,0,0,0,0,0};
  #pragma unroll
  for (int s = 0; s < 4; ++s) {
    int kk = 4 * s + 2 * hi;
    v2f a = *(const v2f*)(X + (size_t)(row0 + n) * 16 + kk);
    v2f b = W1P[s * 32 + lane];
    acc = wmma4(a, b, acc);
  }
  {
    float bb = B1[n];
    #pragma unroll
    for (int r = 0; r < 8; ++r) sm[hi * 8 + r][n] = geluf(acc[r] + bb);
  }
  __builtin_amdgcn_wave_barrier();

  // layer 2
  v8f acc2 = {0,0,0,0,0,0,0,0};
  #pragma unroll
  for (int s = 0; s < 4; ++s) {
    int kk = 4 * s + 2 * hi;
    v2f a; a.x = sm[n][kk]; a.y = sm[n][kk + 1];
    v2f b = W2P[s * 32 + lane];
    acc2 = wmma4(a, b, acc2);
  }
  __builtin_amdgcn_wave_barrier();
  {
    float bb = B2[n];
    #pragma unroll
    for (int r = 0; r < 8; ++r) sm[hi * 8 + r][n] = geluf(acc2[r] + bb);
  }
  __builtin_amdgcn_wave_barrier();

  // layer 3: 16 -> coutPad, no activation
  for (int tile = 0; tile < ntiles; ++tile) {
    v8f acc3 = {0,0,0,0,0,0,0,0};
    #pragma unroll
    for (int s = 0; s < 4; ++s) {
      int kk = 4 * s + 2 * hi;
      v2f a; a.x = sm[n][kk]; a.y = sm[n][kk + 1];
      v2f b = W3P[tile * 128 + s * 32 + lane];
      acc3 = wmma4(a, b, acc3);
    }
    float bb = B3[tile * 16 + n];
    #pragma unroll
    for (int r = 0; r < 8; ++r)
      Y[(size_t)(row0 + hi * 8 + r) * coutPad + tile * 16 + n] =
          (acc3[r] + bb) * scale;
  }
}

// ---------------------------------------------------------------------------
// Generic WMMA GEMM: C[z] = A[z] @ B[z] (+ C[z] if beta), row-major f32.
// Tiles staged via GLOBAL_LOAD_ASYNC_TO_LDS_B128 (ASYNCcnt).
// M,K,Nc multiples of 16; M multiple of 64. grid=(Nc/16, M/64, slabs)
// LDS rows padded to 20 floats (80 B) -> every 16 B chunk stays aligned.
// ---------------------------------------------------------------------------
#define GW 4
__global__ void gemm16(const float* __restrict__ A, long sA,
                       const float* __restrict__ B, long sB,
                       float* __restrict__ C, long sC,
                       int M, int K, int Nc, int beta) {
  __shared__ __align__(16) float As[GW][16][20];
  __shared__ __align__(16) float Bs[16][20];
  int wave = threadIdx.x >> 5;
  int lane = threadIdx.x & 31;
  int n  = lane & 15;
  int hi = lane >> 4;
  int m0 = (blockIdx.y * GW + wave) * 16;
  int n0 = blockIdx.x * 16;
  long z = blockIdx.z;
  const float* Ap = A + z * sA;
  const float* Bp = B + z * sB;
  float*       Cp = C + z * sC;

  int arow = lane >> 2, aq = (lane & 3) * 4;   // 16B chunk per lane
  v8f acc = {0,0,0,0,0,0,0,0};
  for (int kc = 0; kc < K; kc += 16) {
    // A tile (per wave): 16 rows x 64 B = 2 async wave-ops
    async_lds_b128((unsigned)(uintptr_t)&As[wave][arow][aq],
                   Ap + (size_t)(m0 + arow) * K + kc + aq);
    async_lds_b128((unsigned)(uintptr_t)&As[wave][8 + arow][aq],
                   Ap + (size_t)(m0 + 8 + arow) * K + kc + aq);
    // B tile (shared): loaded by wave 0 only
    if (wave == 0) {
      async_lds_b128((unsigned)(uintptr_t)&Bs[arow][aq],
                     Bp + (size_t)(kc + arow) * Nc + n0 + aq);
      async_lds_b128((unsigned)(uintptr_t)&Bs[8 + arow][aq],
                     Bp + (size_t)(kc + 8 + arow) * Nc + n0 + aq);
    }
    wait_async0();          // each wave drains its own async ops
    __syncthreads();        // then LDS contents visible block-wide
    #pragma unroll
    for (int k0 = 0; k0 < 16; k0 += 4) {
      int kk = k0 + 2 * hi;
      v2f a; a.x = As[wave][n][kk]; a.y = As[wave][n][kk + 1];
      v2f b; b.x = Bs[kk][n];       b.y = Bs[kk + 1][n];
      acc = wmma4(a, b, acc);
    }
    __syncthreads();
  }
  #pragma unroll
  for (int r = 0; r < 8; ++r) {
    size_t off = (size_t)(m0 + hi * 8 + r) * Nc + n0 + n;
    float v = acc[r];
    if (beta) v += Cp[off];
    Cp[off] = v;
  }
}

// du = u @ delta (per-pixel 8x8), written planar [n][c][h][w] for the DFTs
__global__ void du_delta(const float* __restrict__ u,
                         const float* __restrict__ delta,
                         float* __restrict__ dupl) {
  int p = blockIdx.x * blockDim.x + threadIdx.x;
  if (p >= PIX) return;
  int w = p & 255, h = (p >> 8) & 255, nb = p >> 16;
  float uj[8];
  #pragma unroll
  for (int j = 0; j < 8; ++j) uj[j] = u[(size_t)p * 16 + j];
  const float* m = delta + (size_t)p * 64;
  #pragma unroll
  for (int k = 0; k < 8; ++k) {
    float a = 0.f;
    #pragma unroll
    for (int j = 0; j < 8; ++j) a += uj[j] * m[j * 8 + k];
    dupl[(((size_t)nb * CC + k) * HH + h) * WWI + w] = a;
  }
}

// u_new = 2u - u@gamma + G@gamma = 2u + (G-u)@gamma ; G read planar
__global__ void combine(const float* __restrict__ u,
                        const float* __restrict__ gamma,
                        const float* __restrict__ gpl,
                        float* __restrict__ un) {
  int p = blockIdx.x * blockDim.x + threadIdx.x;
  if (p >= PIX) return;
  int w = p & 255, h = (p >> 8) & 255, nb = p >> 16;
  float uj[8], tj[8];
  #pragma unroll
  for (int j = 0; j < 8; ++j) {
    uj[j] = u[(size_t)p * 16 + j];
    tj[j] = gpl[(((size_t)nb * CC + j) * HH + h) * WWI + w] - uj[j];
  }
  const float* m = gamma + (size_t)p * 64;
  #pragma unroll
  for (int k = 0; k < 8; ++k) {
    float a = 2.f * uj[k];
    #pragma unroll
    for (int j = 0; j < 8; ++j) a += tj[j] * m[j * 8 + k];
    un[(size_t)p * 16 + k] = a;
  }
  #pragma unroll
  for (int k = 8; k < 16; ++k) un[(size_t)p * 16 + k] = 0.f;
}

__global__ void bias_gelu_k(float* __restrict__ u, const float* __restrict__ bias) {
  int p = blockIdx.x * blockDim.x + threadIdx.x;
  if (p >= PIX) return;
  #pragma unroll
  for (int k = 0; k < 8; ++k)
    u[(size_t)p * 16 + k] = geluf(u[(size_t)p * 16 + k] + bias[k]);
}

// per-frequency complex 8x8: out = uf @ (fr + i*fi)
__global__ void filt_apply(const float* __restrict__ vre, const float* __restrict__ vim,
                           const float* __restrict__ fr,  const float* __restrict__ fi,
                           float* __restrict__ ore, float* __restrict__ oim) {
  int t = blockIdx.x * blockDim.x + threadIdx.x;
  if (t >= NB * HH * FP) return;
  int f = t % FP, g = (t / FP) % HH, nb = t / (HH * FP);
  size_t base = ((size_t)nb * CC) * SLAB + (size_t)g * FP + f;
  float xr[8], xi[8];
  #pragma unroll
  for (int c = 0; c < 8; ++c) {
    xr[c] = vre[base + (size_t)c * SLAB];
    xi[c] = vim[base + (size_t)c * SLAB];
  }
  const float* mr = fr + ((size_t)g * FP + f) * 64;
  const float* mi = fi + ((size_t)g * FP + f) * 64;
  #pragma unroll
  for (int k = 0; k < 8; ++k) {
    float ar = 0.f, ai = 0.f;
    #pragma unroll
    for (int j = 0; j < 8; ++j) {
      ar += xr[j] * mr[j * 8 + k] - xi[j] * mi[j * 8 + k];
      ai += xr[j] * mi[j * 8 + k] + xi[j] * mr[j * 8 + k];
    }
    ore[base + (size_t)k * SLAB] = ar;
    oim[base + (size_t)k * SLAB] = ai;
  }
}

__global__ void extract_out(const float* __restrict__ t, float* __restrict__ out) {
  int p = blockIdx.x * blockDim.x + threadIdx.x;
  if (p < PIX) out[p] = t[(size_t)p * 16];
}

// ---------------------------------------------------------------------------
struct ProjW { const float *w1,*b1,*w2,*b2,*w3,*b3; };

extern "C" void kernel_launch(void* const* d_in, const int* in_sizes, int n_in,
                              void* d_out, int out_size, void* d_ws, size_t ws_size,
                              hipStream_t stream) {
  (void)in_sizes; (void)n_in; (void)out_size; (void)ws_size;
  const float* sos = (const float*)d_in[0];
  const float* pml = (const float*)d_in[1];
  const float* src = (const float*)d_in[2];
  int ii = 3;
  auto getProj = [&](void) -> ProjW {
    ProjW p;
    p.w1 = (const float*)d_in[ii++]; p.b1 = (const float*)d_in[ii++];
    p.w2 = (const float*)d_in[ii++]; p.b2 = (const float*)d_in[ii++];
    p.w3 = (const float*)d_in[ii++]; p.b3 = (const float*)d_in[ii++];
    return p;
  };
  ProjW lift = getProj();
  ProjW gP[DEPTH], dP[DEPTH], grP[DEPTH], giP[DEPTH];
  const float* biasP[DEPTH];
  for (int d = 0; d < DEPTH; ++d) {
    gP[d] = getProj(); dP[d] = getProj();
    grP[d] = getProj(); giP[d] = getProj();
    biasP[d] = (const float*)d_in[ii++];
  }
  ProjW tail = getProj();

  // workspace carve (f32 everywhere)
  size_t off = 0;
  auto carve = [&](size_t bytes) -> float* {
    float* p = (float*)((char*)d_ws + off);
    off += (bytes + 255) & ~(size_t)255;
    return p;
  };
  float* kf   = carve((size_t)PIX * 16 * 4);
  float* kq   = carve((size_t)MFREQ * 16 * 4);
  float* u    = carve((size_t)PIX * 16 * 4);
  float* u2   = carve((size_t)PIX * 16 * 4);
  float* gam  = carve((size_t)PIX * 64 * 4);
  float* del  = carve((size_t)PIX * 64 * 4);
  float* dupl = carve((size_t)MROWS * WWI * 4);
  float* ure  = carve((size_t)NSLAB * SLAB * 4);
  float* uim  = carve((size_t)NSLAB * SLAB * 4);
  float* vre  = carve((size_t)NSLAB * SLAB * 4);
  float* vim  = carve((size_t)NSLAB * SLAB * 4);
  float* fre  = carve((size_t)MFREQ * 64 * 4);
  float* fim  = carve((size_t)MFREQ * 64 * 4);
  float* Bc   = carve((size_t)HH * FP * 4);
  float* Bns  = carve((size_t)HH * FP * 4);
  float* Tc   = carve((size_t)HH * HH * 4);
  float* Ts   = carve((size_t)HH * HH * 4);
  float* Tns  = carve((size_t)HH * HH * 4);
  float* IBc  = carve((size_t)FP * HH * 4);
  float* IBs  = carve((size_t)FP * HH * 4);
  float* pk   = carve((size_t)18 * PKSTRIDE * 4);   // packed weights
  float* ttmp = gam;  // tail output reuses gamma buffer

  // pack all 18 projections into fragment layout
  float* pkLift = pk;
  float* pkG[DEPTH]; float* pkD[DEPTH]; float* pkGR[DEPTH]; float* pkGI[DEPTH];
  float* pkTail = pk + (size_t)17 * PKSTRIDE;
  pack_proj<<<1, 256, 0, stream>>>(lift.w1, lift.b1, 9, lift.w2, lift.b2,
                                   lift.w3, lift.b3, 8, 1, pkLift);
  for (int d = 0; d < DEPTH; ++d) {
    pkG[d]  = pk + (size_t)(1 + d * 4 + 0) * PKSTRIDE;
    pkD[d]  = pk + (size_t)(1 + d * 4 + 1) * PKSTRIDE;
    pkGR[d] = pk + (size_t)(1 + d * 4 + 2) * PKSTRIDE;
    pkGI[d] = pk + (size_t)(1 + d * 4 + 3) * PKSTRIDE;
    pack_proj<<<1, 256, 0, stream>>>(gP[d].w1, gP[d].b1, 9, gP[d].w2, gP[d].b2,
                                     gP[d].w3, gP[d].b3, 64, 4, pkG[d]);
    pack_proj<<<1, 256, 0, stream>>>(dP[d].w1, dP[d].b1, 9, dP[d].w2, dP[d].b2,
                                     dP[d].w3, dP[d].b3, 64, 4, pkD[d]);
    pack_proj<<<1, 256, 0, stream>>>(grP[d].w1, grP[d].b1, 2, grP[d].w2, grP[d].b2,
                                     grP[d].w3, grP[d].b3, 64, 4, pkGR[d]);
    pack_proj<<<1, 256, 0, stream>>>(giP[d].w1, giP[d].b1, 2, giP[d].w2, giP[d].b2,
                                     giP[d].w3, giP[d].b3, 64, 4, pkGI[d]);
  }
  pack_proj<<<1, 256, 0, stream>>>(tail.w1, tail.b1, 8, tail.w2, tail.b2,
                                   tail.w3, tail.b3, 1, 1, pkTail);

  init_bases<<<HH * HH / 256, 256, 0, stream>>>(Bc, Bns, Tc, Ts, Tns, IBc, IBs);
  build_k<<<PIX / 256, 256, 0, stream>>>(sos, pml, src, kf);
  build_kq<<<MFREQ / 256, 256, 0, stream>>>(kq);

  // lift: 9 -> 16 -> 16 -> 8 (stored stride 16, pads zero)
  proj_mlp<<<PIX / (16 * PW), 256, 0, stream>>>(kf, PIX, pkLift, 1, u, 1.0f);

  float* ucur = u;
  float* ualt = u2;

  for (int d = 0; d < DEPTH; ++d) {
    proj_mlp<<<PIX / (16 * PW), 256, 0, stream>>>(kf, PIX, pkG[d], 4, gam,
                                                  1.0f / 64.f);
    proj_mlp<<<PIX / (16 * PW), 256, 0, stream>>>(kf, PIX, pkD[d], 4, del,
                                                  1.0f / 64.f);
    proj_mlp<<<MFREQ / (16 * PW), 256, 0, stream>>>(kq, MFREQ, pkGR[d], 4, fre,
                                                    1.0f / 64.f);
    proj_mlp<<<MFREQ / (16 * PW), 256, 0, stream>>>(kq, MFREQ, pkGI[d], 4, fim,
                                                    1.0f / 64.f);

    for (int it = 0; it < ITERS; ++it) {
      du_delta<<<PIX / 256, 256, 0, stream>>>(ucur, del, dupl);

      // forward rFFT along W as GEMM: [8192x256] @ [256x144]
      dim3 gW(FP / 16, MROWS / (16 * GW), 1);
      gemm16<<<gW, 128, 0, stream>>>(dupl, 0, Bc, 0, ure, 0, MROWS, WWI, FP, 0);
      gemm16<<<gW, 128, 0, stream>>>(dupl, 0, Bns, 0, uim, 0, MROWS, WWI, FP, 0);

      // forward complex FFT along H, per (n,c) slab: [256x256] @ [256x144]
      dim3 gH(FP / 16, HH / (16 * GW), NSLAB);
      gemm16<<<gH, 128, 0, stream>>>(Tc, 0, ure, SLAB, vre, SLAB, HH, HH, FP, 0);
      gemm16<<<gH, 128, 0, stream>>>(Ts, 0, uim, SLAB, vre, SLAB, HH, HH, FP, 1);
      gemm16<<<gH, 128, 0, stream>>>(Tc, 0, uim, SLAB, vim, SLAB, HH, HH, FP, 0);
      gemm16<<<gH, 128, 0, stream>>>(Tns, 0, ure, SLAB, vim, SLAB, HH, HH, FP, 1);

      // per-frequency complex 8x8 filter
      filt_apply<<<(NB * HH * FP) / 256, 256, 0, stream>>>(vre, vim, fre, fim,
                                                           ure, uim);

      // inverse FFT along H (e^{+i}): Xre=Tc@P_re+Tns@P_im, Xim=Ts@P_re+Tc@P_im
      gemm16<<<gH, 128, 0, stream>>>(Tc, 0, ure, SLAB, vre, SLAB, HH, HH, FP, 0);
      gemm16<<<gH, 128, 0, stream>>>(Tns, 0, uim, SLAB, vre, SLAB, HH, HH, FP, 1);
      gemm16<<<gH, 128, 0, stream>>>(Ts, 0, ure, SLAB, vim, SLAB, HH, HH, FP, 0);
      gemm16<<<gH, 128, 0, stream>>>(Tc, 0, uim, SLAB, vim, SLAB, HH, HH, FP, 1);

      // inverse rFFT along W (Hermitian, normalized): G = Xre@IBc + Xim@IBs
      dim3 gIW(WWI / 16, MROWS / (16 * GW), 1);
      gemm16<<<gIW, 128, 0, stream>>>(vre, 0, IBc, 0, dupl, 0, MROWS, FP, WWI, 0);
      gemm16<<<gIW, 128, 0, stream>>>(vim, 0, IBs, 0, dupl, 0, MROWS, FP, WWI, 1);

      // u <- 2u + (G - u) @ gamma
      combine<<<PIX / 256, 256, 0, stream>>>(ucur, gam, dupl, ualt);
      float* tmp = ucur; ucur = ualt; ualt = tmp;
    }
    bias_gelu_k<<<PIX / 256, 256, 0, stream>>>(ucur, biasP[d]);
  }

  // tail: 8 -> 16 -> 16 -> 1
  proj_mlp<<<PIX / (16 * PW), 256, 0, stream>>>(ucur, PIX, pkTail, 1, ttmp, 1.0f);
  extract_out<<<PIX / 256, 256, 0, stream>>>(ttmp, (float*)d_out);
}